// AttentionMV_41257455846090
// MI455X (gfx1250) — compile-verified
//
#include <hip/hip_runtime.h>
#include <hip/hip_bf16.h>
#include <math.h>

#define B_  64
#define T_  2048
#define E_  1024
#define CH_ 16            // T-chunks per batch row
#define TC_ (T_ / CH_)    // 128 timesteps per chunk

typedef __attribute__((ext_vector_type(2))) float v2f;
typedef __attribute__((ext_vector_type(8))) float v8f;

// ---------------------------------------------------------------------------
// Kernel 1: et_base[b,t] = sum_e c[b,e] * U[e,t]
// f32 GEMM M=64, N=2048, K=1024 via V_WMMA_F32_16X16X4_F32 (one 16x16 tile
// per wave32, K stepped by 4). Layouts per CDNA5 ISA 7.12.2:
//   A 16x4 f32 : lanes 0-15 -> M=lane,  VGPR0=K0, VGPR1=K1
//                lanes16-31 -> M=lane-16,VGPR0=K2, VGPR1=K3
//   B 4x16 f32 : lanes 0-15 -> N=lane,  VGPR0=K0, VGPR1=K1
//                lanes16-31 -> N=lane-16,VGPR0=K2, VGPR1=K3
//   C/D 16x16  : VGPR r: lanes 0-15 -> M=r, lanes 16-31 -> M=r+8; N=lane&15
// ---------------------------------------------------------------------------
__global__ void __launch_bounds__(32)
etbase_wmma_kernel(const float* __restrict__ c, const float* __restrict__ U,
                   float* __restrict__ et) {
  const int lane = threadIdx.x;
  const int half = lane >> 4;     // 0 or 1
  const int l16  = lane & 15;
  const int m0   = blockIdx.x * 16;   // 0..48
  const int n0   = blockIdx.y * 16;   // 0..2032

  const float* __restrict__ crow = c + (size_t)(m0 + l16) * E_; // A row
  const float* __restrict__ ucol = U + (size_t)(n0 + l16);      // B col base

  v8f acc = {};
  for (int kk = 0; kk < E_; kk += 4) {
    v2f a;
    a.x = crow[kk + half * 2 + 0];
    a.y = crow[kk + half * 2 + 1];
    v2f b;
    b.x = ucol[(size_t)(kk + half * 2 + 0) * T_];
    b.y = ucol[(size_t)(kk + half * 2 + 1) * T_];
    acc = __builtin_amdgcn_wmma_f32_16x16x4_f32(
        /*neg_a=*/false, a, /*neg_b=*/false, b,
        /*c_mod=*/(short)0, acc, /*reuse_a=*/false, /*reuse_b=*/false);
  }
#pragma unroll
  for (int r = 0; r < 8; ++r) {
    const int row = m0 + r + half * 8;
    et[(size_t)row * T_ + n0 + l16] = acc[r];
  }
}

// ---------------------------------------------------------------------------
// Kernel 2: one wave32 per (batch, T-chunk). Online softmax over the chunk,
// fused with the weighted accumulation so x is read from HBM exactly once.
// Lane owns E-slice e = lane*4 + j*128 (8 x float4 in registers).
// ---------------------------------------------------------------------------
__global__ void __launch_bounds__(32)
chunk_online_softmax_kernel(const float* __restrict__ x,
                            const float* __restrict__ W,
                            const float* __restrict__ bvec,
                            const float* __restrict__ et,
                            float* __restrict__ pm, float* __restrict__ ps,
                            float* __restrict__ pacc) {
  const int gb   = blockIdx.x;      // 0 .. B*CH-1
  const int b    = gb / CH_;
  const int ch   = gb % CH_;
  const int lane = threadIdx.x;
  const int t0   = ch * TC_;

  // Preload this lane's slice of W (constant across all timesteps).
  float4 wv[8];
#pragma unroll
  for (int j = 0; j < 8; ++j)
    wv[j] = *(const float4*)(W + lane * 4 + j * 128);

  float4 acc[8];
#pragma unroll
  for (int j = 0; j < 8; ++j) acc[j] = make_float4(0.f, 0.f, 0.f, 0.f);

  float m = -INFINITY;
  float s = 0.f;

  const float* __restrict__ xb = x + ((size_t)b * T_ + t0) * E_;
  const float* __restrict__ eb = et + (size_t)b * T_ + t0;

  for (int t = 0; t < TC_; ++t) {
    const float* __restrict__ xr = xb + (size_t)t * E_;
    if (t + 2 < TC_)  // global_prefetch_b8 two rows ahead
      __builtin_prefetch(xr + 2 * (size_t)E_ + lane * 4, 0, 0);

    float4 row[8];
    float d = 0.f;
#pragma unroll
    for (int j = 0; j < 8; ++j) {
      row[j] = *(const float4*)(xr + lane * 4 + j * 128);
      d = fmaf(row[j].x, wv[j].x, d);
      d = fmaf(row[j].y, wv[j].y, d);
      d = fmaf(row[j].z, wv[j].z, d);
      d = fmaf(row[j].w, wv[j].w, d);
    }
    // wave32 butterfly all-reduce of the dot partial
#pragma unroll
    for (int off = 16; off > 0; off >>= 1)
      d += __shfl_xor(d, off, 32);

    const float e  = eb[t] + bvec[t0 + t] + d;   // et[b,t]
    const float nm = fmaxf(m, e);
    const float corr = __expf(m - nm);           // 0 on first step (m=-inf)
    const float p    = __expf(e - nm);
    s = fmaf(s, corr, p);
#pragma unroll
    for (int j = 0; j < 8; ++j) {
      acc[j].x = fmaf(acc[j].x, corr, p * row[j].x);
      acc[j].y = fmaf(acc[j].y, corr, p * row[j].y);
      acc[j].z = fmaf(acc[j].z, corr, p * row[j].z);
      acc[j].w = fmaf(acc[j].w, corr, p * row[j].w);
    }
    m = nm;
  }

  if (lane == 0) { pm[gb] = m; ps[gb] = s; }
  float* __restrict__ pa = pacc + (size_t)gb * E_;
#pragma unroll
  for (int j = 0; j < 8; ++j)
    *(float4*)(pa + lane * 4 + j * 128) = acc[j];
}

// ---------------------------------------------------------------------------
// Kernel 3: merge the CH_ chunk partials per batch (split-softmax combine)
// and write out[b,e] = (sum_i acc_i[e] * exp(m_i - M)) / (sum_i s_i*exp(m_i-M))
// ---------------------------------------------------------------------------
__global__ void __launch_bounds__(256)
combine_kernel(const float* __restrict__ pm, const float* __restrict__ ps,
               const float* __restrict__ pacc, float* __restrict__ out) {
  const int b   = blockIdx.x;
  const int tid = threadIdx.x;

  float M = -INFINITY;
#pragma unroll
  for (int i = 0; i < CH_; ++i) M = fmaxf(M, pm[b * CH_ + i]);

  float w[CH_];
  float S = 0.f;
#pragma unroll
  for (int i = 0; i < CH_; ++i) {
    w[i] = __expf(pm[b * CH_ + i] - M);
    S = fmaf(ps[b * CH_ + i], w[i], S);
  }
  const float inv = 1.f / S;

  for (int e = tid; e < E_; e += 256) {
    float a = 0.f;
#pragma unroll
    for (int i = 0; i < CH_; ++i)
      a = fmaf(pacc[((size_t)(b * CH_ + i)) * E_ + e], w[i], a);
    out[(size_t)b * E_ + e] = a * inv;
  }
}

// ---------------------------------------------------------------------------
extern "C" void kernel_launch(void* const* d_in, const int* in_sizes, int n_in,
                              void* d_out, int out_size, void* d_ws, size_t ws_size,
                              hipStream_t stream) {
  (void)in_sizes; (void)n_in; (void)out_size; (void)ws_size;

  const float* x    = (const float*)d_in[0];  // (B,T,E)
  const float* c    = (const float*)d_in[1];  // (B,E)
  const float* W    = (const float*)d_in[2];  // (E,1)
  const float* bvec = (const float*)d_in[3];  // (T,1)
  const float* U    = (const float*)d_in[4];  // (E,T)
  float* out = (float*)d_out;                 // (B,E)

  // workspace layout (floats): et_base[B*T] | pm[B*CH] | ps[B*CH] | pacc[B*CH*E]
  float* et   = (float*)d_ws;
  float* pm   = et + (size_t)B_ * T_;
  float* ps   = pm + (size_t)B_ * CH_;
  float* pacc = ps + (size_t)B_ * CH_;

  dim3 g1(B_ / 16, T_ / 16);  // 4 x 128 tiles, one wave each
  etbase_wmma_kernel<<<g1, 32, 0, stream>>>(c, U, et);

  chunk_online_softmax_kernel<<<B_ * CH_, 32, 0, stream>>>(x, W, bvec, et,
                                                           pm, ps, pacc);

  combine_kernel<<<B_, 256, 0, stream>>>(pm, ps, pacc, out);
}